// CSA_64106681860574
// MI455X (gfx1250) — compile-verified
//
#include <hip/hip_runtime.h>

typedef float v2f __attribute__((ext_vector_type(2)));
typedef float v8f __attribute__((ext_vector_type(8)));

// ---------------- problem constants ----------------
constexpr int B_ = 2, H_ = 256, W_ = 256, C_ = 256;
constexpr int KK = 9, HEADS = 8, HD = 32;
constexpr int hP = 128, wP = 128, L_ = hP * wP;   // pooled spatial
constexpr int BL = B_ * L_;                        // 32768 GEMM rows
constexpr int ATT = 648;                           // K^4 * heads
constexpr int NPIX = B_ * H_ * W_;                 // 131072

// ---------------- workspace layout ----------------
constexpr size_t alignup(size_t v) { return (v + 255) & ~(size_t)255; }
constexpr size_t OFF_A0   = 0;                                          // pooled (BL,256) f32
constexpr size_t OFF_ARAW = alignup(OFF_A0   + (size_t)BL * C_ * 4);    // attn (BL,648) f32
constexpr size_t OFF_RM   = alignup(OFF_ARAW + (size_t)BL * ATT * 4);   // rowmap 9*BL i32
constexpr size_t OFF_XT   = alignup(OFF_RM   + (size_t)KK * BL * 4);    // XT (B,C,H,W) f32 ; reused as Y0
constexpr size_t OFF_V2   = alignup(OFF_XT   + (size_t)B_ * C_ * H_ * W_ * 4); // V2 (BL,9,256)
constexpr size_t OFF_O    = alignup(OFF_V2   + (size_t)BL * KK * C_ * 4);      // O  (BL,9,256)

// Async global->LDS copy of 16B per lane (CDNA5 ASYNCcnt path).
// lds_off: LDS byte address (low 32 bits of the generic shared pointer,
// per the flat->LDS aperture mapping). gaddr: 64-bit global address.
__device__ __forceinline__ void async_copy_b128(unsigned lds_off, unsigned long long gaddr) {
  asm volatile("global_load_async_to_lds_b128 %0, %1, off"
               :: "v"(lds_off), "v"(gaddr) : "memory");
}
__device__ __forceinline__ void wait_async0() {
  asm volatile("s_wait_asynccnt 0x0" ::: "memory");
}
__device__ __forceinline__ unsigned lds_off_of(const void* p) {
  return (unsigned)(unsigned long long)p;   // low 32 bits = LDS byte offset
}

// ================= 1. avg pool (s=2) =================
__global__ __launch_bounds__(256)
void pool_kernel(const float* __restrict__ x, float* __restrict__ A0) {
  int idx = blockIdx.x * blockDim.x + threadIdx.x;
  const int total = BL * (C_ / 4);
  if (idx >= total) return;
  int c4 = idx & 63;
  int t = idx >> 6;
  int j = t & (wP - 1); t >>= 7;
  int i = t & (hP - 1);
  int b = t >> 7;
  const float* base = x + ((size_t)((b * H_ + 2 * i) * W_ + 2 * j)) * C_ + c4 * 4;
  float s0 = 0.f, s1 = 0.f, s2 = 0.f, s3 = 0.f;
#pragma unroll
  for (int di = 0; di < 2; ++di)
#pragma unroll
    for (int dj = 0; dj < 2; ++dj) {
      const float4 v = *reinterpret_cast<const float4*>(base + ((size_t)di * W_ + dj) * C_);
      s0 += v.x; s1 += v.y; s2 += v.z; s3 += v.w;
    }
  float4 o; o.x = 0.25f * s0; o.y = 0.25f * s1; o.z = 0.25f * s2; o.w = 0.25f * s3;
  *reinterpret_cast<float4*>(A0 + (size_t)(b * L_ + i * wP + j) * C_ + c4 * 4) = o;
}

// ================= 2. WMMA fp32 GEMM, double-buffered async LDS staging =================
// C[m, n] = sum_k Arow(m)[k] * Bw[n, k] (+ bias[n]) ; Bw row-major (N, K)
// Arow(m): rowmap ? (rowmap[m] < 0 ? zeros : A + rowmap[m]) : A + m*K
// M must be a multiple of 64; N guarded; K multiple of 32.
constexpr int TM = 64, TN = 64, KB = 32, LDT = 36;  // LDT: conflict-free LDS stride

__global__ __launch_bounds__(256, 2)
void gemm_wmma_f32(const float* __restrict__ A,
                   const float* __restrict__ Bw,
                   const float* __restrict__ bias,
                   float* __restrict__ Cmat,
                   int M, int N, int K, int ldc,
                   const int* __restrict__ rowmap)
{
  __shared__ float As[2][TM * LDT];
  __shared__ float Bs[2][TN * LDT];
  const int tid  = threadIdx.x;
  const int wave = tid >> 5;
  const int lane = tid & 31;
  const int m0 = blockIdx.x * TM;
  const int n0 = blockIdx.y * TN;
  const int wm = (wave & 3) * 16;      // wave's M sub-tile
  const int wn = (wave >> 2) * 32;     // wave's N sub-strip (two 16-wide tiles)
  const int lm = lane & 15;
  const int hi = lane >> 4;

  // per-thread staging descriptors: 2 float4 slots each for A and B tiles
  int r_[2], cq_[2];
  const float* gA[2]; bool vA[2];
  const float* gB[2]; bool vB[2];
#pragma unroll
  for (int it = 0; it < 2; ++it) {
    int fid = tid + it * 256;          // 0..511 float4 slots
    int r   = fid >> 3;                // tile row (0..63)
    int cq  = (fid & 7) * 4;           // col (float index)
    r_[it] = r; cq_[it] = cq;
    if (rowmap) {
      int rm = rowmap[m0 + r];
      vA[it] = (rm >= 0);
      gA[it] = A + (size_t)(vA[it] ? rm : 0) + cq;
    } else {
      vA[it] = true;
      gA[it] = A + (size_t)(m0 + r) * K + cq;
    }
    int nrow = n0 + r;
    vB[it] = (nrow < N);
    gB[it] = Bw + (size_t)(vB[it] ? nrow : 0) * K + cq;
  }

  auto stage = [&](int buf, int k0) {
#pragma unroll
    for (int it = 0; it < 2; ++it) {
      float* ap = &As[buf][r_[it] * LDT + cq_[it]];
      float* bp = &Bs[buf][r_[it] * LDT + cq_[it]];
      if (vA[it]) async_copy_b128(lds_off_of(ap), (unsigned long long)(const void*)(gA[it] + k0));
      else        *reinterpret_cast<float4*>(ap) = make_float4(0.f, 0.f, 0.f, 0.f);
      if (vB[it]) async_copy_b128(lds_off_of(bp), (unsigned long long)(const void*)(gB[it] + k0));
      else        *reinterpret_cast<float4*>(bp) = make_float4(0.f, 0.f, 0.f, 0.f);
    }
  };

  v8f acc0 = {0.f, 0.f, 0.f, 0.f, 0.f, 0.f, 0.f, 0.f};
  v8f acc1 = {0.f, 0.f, 0.f, 0.f, 0.f, 0.f, 0.f, 0.f};

  stage(0, 0);
  wait_async0();
  __syncthreads();

  const int nblk = K / KB;
  for (int t = 0; t < nblk; ++t) {
    const int cur = t & 1;
    if (t + 1 < nblk) stage(cur ^ 1, (t + 1) * KB);   // overlap next tile with compute

    const float* as = As[cur];
    const float* bs = Bs[cur];
#pragma unroll
    for (int k4 = 0; k4 < KB; k4 += 4) {
      // A 16x4 frag: lanes 0-15 hold K={0,1}, lanes 16-31 hold K={2,3}
      v2f af = *reinterpret_cast<const v2f*>(&as[(wm + lm) * LDT + k4 + hi * 2]);
      v2f b0 = *reinterpret_cast<const v2f*>(&bs[(wn      + lm) * LDT + k4 + hi * 2]);
      v2f b1 = *reinterpret_cast<const v2f*>(&bs[(wn + 16 + lm) * LDT + k4 + hi * 2]);
      acc0 = __builtin_amdgcn_wmma_f32_16x16x4_f32(false, af, false, b0, (short)0, acc0, false, false);
      acc1 = __builtin_amdgcn_wmma_f32_16x16x4_f32(false, af, false, b1, (short)0, acc1, false, false);
    }
    wait_async0();      // next tile fully landed in LDS
    __syncthreads();    // all waves done reading 'cur' / producers done
  }

  // C/D layout: VGPR v holds row (v + 8*hi), lane&15 is the column
  const int cn = lane & 15;
  const int mb = wm + hi * 8;
  const int n1 = n0 + wn + cn;
  const int n2 = n1 + 16;
  const float bb1 = (bias && n1 < N) ? bias[n1] : 0.f;
  const float bb2 = (bias && n2 < N) ? bias[n2] : 0.f;
#pragma unroll
  for (int v = 0; v < 8; ++v) {
    int m = m0 + mb + v;
    if (n1 < N) Cmat[(size_t)m * ldc + n1] = acc0[v] + bb1;
    if (n2 < N) Cmat[(size_t)m * ldc + n2] = acc1[v] + bb2;
  }
}

// ================= 3. softmax over trailing 9 (with 1/sqrt(hd) scale) =================
__global__ __launch_bounds__(256)
void softmax_kernel(float* __restrict__ A) {
  int idx = blockIdx.x * blockDim.x + threadIdx.x;
  const int total = BL * HEADS * KK;
  if (idx >= total) return;
  int bl = idx / (HEADS * KK);
  int r  = idx - bl * (HEADS * KK);
  float* row = A + (size_t)bl * ATT + r * KK;
  const float scale = 0.17677669529663687f;  // 32^-0.5
  float v[KK];
  float mx = -3.4e38f;
#pragma unroll
  for (int q = 0; q < KK; ++q) { v[q] = row[q] * scale; mx = fmaxf(mx, v[q]); }
  float s = 0.f;
#pragma unroll
  for (int q = 0; q < KK; ++q) { v[q] = __expf(v[q] - mx); s += v[q]; }
  float inv = 1.f / s;
#pragma unroll
  for (int q = 0; q < KK; ++q) row[q] = v[q] * inv;
}

// ================= 4. x (B,H,W,C) -> XT (B,C,H,W) =================
__global__ void transpose_kernel(const float* __restrict__ x, float* __restrict__ XT) {
  __shared__ float tile[32][33];
  int bh = blockIdx.z;                // b*H + h
  int b = bh >> 8, h = bh & 255;
  const float* src = x + (size_t)bh * (W_ * C_);
  int w0 = blockIdx.x * 32, c0 = blockIdx.y * 32;
#pragma unroll
  for (int k = 0; k < 4; ++k)
    tile[threadIdx.y + k * 8][threadIdx.x] =
        src[(size_t)(w0 + threadIdx.y + k * 8) * C_ + c0 + threadIdx.x];
  __syncthreads();
#pragma unroll
  for (int k = 0; k < 4; ++k)
    XT[((size_t)(b * C_ + c0 + threadIdx.y + k * 8) * H_ + h) * W_ + w0 + threadIdx.x] =
        tile[threadIdx.x][threadIdx.y + k * 8];
}

// ================= 5. gather-row offsets for the 9 unfold-GEMMs =================
// row (b,i,j) of patch p=(a,bb): XT[b, 2i+a-1, 2j+bb-1, :]  (-1 => zero row)
__global__ __launch_bounds__(256)
void rowmap_kernel(int* __restrict__ rowmap) {
  int idx = blockIdx.x * blockDim.x + threadIdx.x;
  if (idx >= KK * BL) return;
  int p = idx >> 15;           // BL == 2^15
  int m = idx & (BL - 1);
  int a = p / 3, bb = p - a * 3;
  int b = m >> 14;
  int i = (m >> 7) & 127;
  int j = m & 127;
  int cp = 2 * i + a - 1;
  int hp = 2 * j + bb - 1;
  rowmap[idx] = (cp >= 0 && cp < C_ && hp >= 0 && hp < H_)
                    ? ((b * C_ + cp) * H_ + hp) * W_
                    : -1;
}

// ================= 6. attention apply: per (b,l,head): (9x9)@(9x32) =================
__global__ __launch_bounds__(256)
void apply_kernel(const float* __restrict__ Asm, const float* __restrict__ V2,
                  float* __restrict__ O) {
  int wid  = blockIdx.x * 8 + (threadIdx.x >> 5);
  int lane = threadIdx.x & 31;
  if (wid >= BL * HEADS * KK) return;
  int bl = wid / (HEADS * KK);
  int np = wid - bl * (HEADS * KK);     // n*9 + p
  int n  = np / KK;
  int p  = np - n * KK;
  const float* arow = Asm + (size_t)bl * ATT + np * KK;        // = bl*648 + n*81 + p*9
  const float* vb   = V2  + (size_t)bl * (KK * C_) + n * HD + lane;
  float acc = 0.f;
#pragma unroll
  for (int q = 0; q < KK; ++q) acc = fmaf(arow[q], vb[(size_t)q * C_], acc);
  O[(size_t)bl * (KK * C_) + p * C_ + n * HD + lane] = acc;
}

// ================= 7. fold: gather-add (<=4 contributions / output pixel) =================
__global__ __launch_bounds__(256)
void fold_kernel(const float* __restrict__ O, float* __restrict__ Y0) {
  int idx = blockIdx.x * blockDim.x + threadIdx.x;
  const int total = NPIX * (C_ / 4);
  if (idx >= total) return;
  int c4 = idx & 63;
  int t = idx >> 6;
  int ww = t & 255; t >>= 8;
  int hh = t & 255;
  int b  = t >> 8;
  float s0 = 0.f, s1 = 0.f, s2 = 0.f, s3 = 0.f;
#pragma unroll
  for (int ki = 0; ki < 3; ++ki) {
    int ni = hh + 1 - ki;
    if (ni < 0 || (ni & 1)) continue;
    int i = ni >> 1; if (i >= hP) continue;
#pragma unroll
    for (int kj = 0; kj < 3; ++kj) {
      int nj = ww + 1 - kj;
      if (nj < 0 || (nj & 1)) continue;
      int j = nj >> 1; if (j >= wP) continue;
      const float4 v = *reinterpret_cast<const float4*>(
          O + ((size_t)(b * L_ + i * wP + j) * KK + ki * 3 + kj) * C_ + c4 * 4);
      s0 += v.x; s1 += v.y; s2 += v.z; s3 += v.w;
    }
  }
  float4 o; o.x = s0; o.y = s1; o.z = s2; o.w = s3;
  *reinterpret_cast<float4*>(Y0 + (size_t)((b * H_ + hh) * W_ + ww) * C_ + c4 * 4) = o;
}

// ================= host-side orchestration =================
extern "C" void kernel_launch(void* const* d_in, const int* in_sizes, int n_in,
                              void* d_out, int out_size, void* d_ws, size_t ws_size,
                              hipStream_t stream) {
  (void)in_sizes; (void)n_in; (void)out_size; (void)ws_size;
  const float* x      = (const float*)d_in[0];
  const float* attn_w = (const float*)d_in[1];
  const float* attn_b = (const float*)d_in[2];
  const float* conv_w = (const float*)d_in[3];
  const float* proj_w = (const float*)d_in[4];
  const float* proj_b = (const float*)d_in[5];
  float* out = (float*)d_out;

  char* ws = (char*)d_ws;
  float* A0   = (float*)(ws + OFF_A0);
  float* ARAW = (float*)(ws + OFF_ARAW);
  int*   RM   = (int*)  (ws + OFF_RM);
  float* XT   = (float*)(ws + OFF_XT);
  float* Y0   = XT;  // XT dead after the conv GEMMs; reuse for folded output
  float* V2   = (float*)(ws + OFF_V2);
  float* Obuf = (float*)(ws + OFF_O);

  // 1) pool
  { int total = BL * (C_ / 4);
    pool_kernel<<<(total + 255) / 256, 256, 0, stream>>>(x, A0); }

  // 2) attention logits GEMM: (BL,256)x(256,648)+bias
  { dim3 g(BL / TM, (ATT + TN - 1) / TN);
    gemm_wmma_f32<<<g, 256, 0, stream>>>(A0, attn_w, attn_b, ARAW, BL, ATT, C_, ATT, nullptr); }

  // 3) scale + softmax (groups of 9), in place
  { int total = BL * HEADS * KK;
    softmax_kernel<<<(total + 255) / 256, 256, 0, stream>>>(ARAW); }

  // 4) transpose x -> XT (contraction axis of unfold-GEMM made contiguous)
  { dim3 g(8, 8, B_ * H_); dim3 blk(32, 8);
    transpose_kernel<<<g, blk, 0, stream>>>(x, XT); }

  // 5) gather row offsets
  { int total = KK * BL;
    rowmap_kernel<<<(total + 255) / 256, 256, 0, stream>>>(RM); }

  // 6) 9 gathered GEMMs: V2[b,l,p,:] = XT_row x conv_w[p]^T
  for (int p = 0; p < KK; ++p) {
    dim3 g(BL / TM, C_ / TN);
    gemm_wmma_f32<<<g, 256, 0, stream>>>(XT, conv_w + (size_t)p * C_ * C_, nullptr,
                                         V2 + (size_t)p * C_, BL, C_, C_, KK * C_,
                                         RM + p * BL);
  }

  // 7) attention apply
  { int nwaves = BL * HEADS * KK;
    apply_kernel<<<nwaves / 8, 256, 0, stream>>>(ARAW, V2, Obuf); }

  // 8) fold (deterministic gather)
  { int total = NPIX * (C_ / 4);
    fold_kernel<<<(total + 255) / 256, 256, 0, stream>>>(Obuf, Y0); }

  // 9) projection GEMM: (131072,256)x(256,256)+bias -> out (B,H,W,C)
  { dim3 g(NPIX / TM, C_ / TN);
    gemm_wmma_f32<<<g, 256, 0, stream>>>(Y0, proj_w, proj_b, out, NPIX, C_, C_, C_, nullptr); }
}